// MyModel_61933428411406
// MI455X (gfx1250) — compile-verified
//
#include <hip/hip_runtime.h>

typedef __attribute__((ext_vector_type(2))) float v2f;
typedef __attribute__((ext_vector_type(8))) float v8f;

#define WAVES_PER_BLOCK 8
#define BLOCK_THREADS (WAVES_PER_BLOCK * 32)

__device__ __forceinline__ int lower_bound_i32(const int* __restrict__ a, int n, int key) {
  int lo = 0, hi = n;
  while (lo < hi) {
    int mid = (lo + hi) >> 1;
    int v = a[mid];
    if (v < key) lo = mid + 1; else hi = mid;
  }
  return lo;
}

// Process 4 edges [e, e+4) with one K=4 WMMA step per 16-wide feature block.
// GUARD=false: caller guarantees e+4 <= e_end (hot path, no per-lane compares).
template <bool GUARD>
__device__ __forceinline__ void spmm_step4(const float* __restrict__ x,
                                           const float* __restrict__ vals,
                                           const int*   __restrict__ row,
                                           const int*   __restrict__ col,
                                           int e, int e_end, int row_base,
                                           int kb, int fn,
                                           v8f& acc0, v8f& acc1, v8f& acc2, v8f& acc3)
{
  const int e0 = e + kb;
  const int e1 = e0 + 1;
  int sel0 = 1, sel1 = 1;
  int ce0 = e0, ce1 = e1;
  if (GUARD) {
    sel0 = (e0 < e_end);
    sel1 = (e1 < e_end);
    ce0 = sel0 ? e0 : 0;
    ce1 = sel1 ? e1 : 0;
  }

  // Consecutive indices -> compiler merges each pair into global_load_b64.
  const int   r0 = row[ce0], r1 = row[ce1];
  const int   c0 = col[ce0], c1 = col[ce1];
  const float w0 = vals[ce0], w1 = vals[ce1];

  // A selection fragment: vals[e+k] lands on its owning local row, else 0.
  // Non-short-circuit & keeps the loads unconditional (no exec-mask branch).
  v2f a;
  a.x = ((sel0 & (r0 == row_base + fn)) != 0) ? w0 : 0.0f;
  a.y = ((sel1 & (r1 == row_base + fn)) != 0) ? w1 : 0.0f;

  // B fragments: gathered feature rows (x = 25.6MB -> L2-resident).
  // 32-bit unsigned element offsets (x < 4GB) so the backend emits the GVS
  // saddr + 32-bit-voffset form instead of 64-bit VALU address math.
  const unsigned o0 = (unsigned)c0 * 64u + (unsigned)fn;
  const unsigned o1 = (unsigned)c1 * 64u + (unsigned)fn;
  v2f b0; b0.x = x[o0];       b0.y = x[o1];
  v2f b1; b1.x = x[o0 + 16u]; b1.y = x[o1 + 16u];
  v2f b2; b2.x = x[o0 + 32u]; b2.y = x[o1 + 32u];
  v2f b3; b3.x = x[o0 + 48u]; b3.y = x[o1 + 48u];

  acc0 = __builtin_amdgcn_wmma_f32_16x16x4_f32(false, a, false, b0, (short)0, acc0, false, false);
  acc1 = __builtin_amdgcn_wmma_f32_16x16x4_f32(false, a, false, b1, (short)0, acc1, false, false);
  acc2 = __builtin_amdgcn_wmma_f32_16x16x4_f32(false, a, false, b2, (short)0, acc2, false, false);
  acc3 = __builtin_amdgcn_wmma_f32_16x16x4_f32(false, a, false, b3, (short)0, acc3, false, false);
}

// One wave32 per 16-row group. Sorted-COO rows => each group's edges are a
// contiguous range [e_start, e_end). Segmented sum over that range is done with
// V_WMMA_F32_16X16X4_F32: A is a 16x4 "selection" matrix carrying vals[e] at
// (local_row, k); B holds 4 gathered feature rows x[col[e]] per 16-wide block;
// C accumulates the 16x16 fp32 output tiles (precision matches the reference).
__global__ __launch_bounds__(BLOCK_THREADS)
void spmm_mean_wmma_f32(const float* __restrict__ x,
                        const float* __restrict__ vals,
                        const int*   __restrict__ row,
                        const int*   __restrict__ col,
                        float*       __restrict__ out,
                        int n_nodes, int n_edges)
{
  const int lane = (int)(threadIdx.x & 31u);
  const int wid  = (int)(blockIdx.x * WAVES_PER_BLOCK + (threadIdx.x >> 5));
  // Wave-uniform guard via scalar compare: EXEC is never partially masked.
  const int row_base = __builtin_amdgcn_readfirstlane(wid << 4);
  if (row_base >= n_nodes) return;

  // ---- lane-parallel CSR boundary search: lanes 0..16 find the 17 segment
  //      boundaries; per-row degrees are adjacent differences. ----
  const int tgt = row_base + (lane < 16 ? lane : 16);
  const int b   = lower_bound_i32(row, n_edges, tgt);
  // Loop bounds forced into SGPRs -> uniform (s_cmp/s_cbranch) edge loop.
  const int e_start = __builtin_amdgcn_readfirstlane(__shfl(b, 0, 32));
  const int e_end   = __builtin_amdgcn_readfirstlane(__shfl(b, 16, 32));
  const int b_next  = __shfl(b, (lane + 1) & 31, 32);
  const int deg     = b_next - b;                         // valid in lanes 0..15
  const float rdeg  = 1.0f / (float)((deg < 1) ? 1 : deg);

  // WMMA fragment geometry (wave32):
  //   A 16x4 f32 : lanes 0-15 -> M=lane, K=0,1 ; lanes 16-31 -> K=2,3
  //   B 4x16 f32 : VGPR0 = K=(0|2), VGPR1 = K=(1|3), striped over N=lane&15
  //   C 16x16    : 8 VGPRs, element (M = v + 8*(lane>=16), N = lane&15)
  const int half = lane >> 4;
  const int kb   = half << 1;
  const int fn   = lane & 15;

  v8f acc0 = {}, acc1 = {}, acc2 = {}, acc3 = {};

  int e = e_start;
  // Hot path: 8 edges (two K=4 steps) per iteration, no per-lane guards;
  // ~16 outstanding gathers in flight for latency hiding.
  for (; e + 8 <= e_end; e += 8) {
    __builtin_prefetch(row  + e + 256, 0, 3);
    __builtin_prefetch(col  + e + 256, 0, 3);
    __builtin_prefetch(vals + e + 256, 0, 3);
    spmm_step4<false>(x, vals, row, col, e,     e_end, row_base, kb, fn, acc0, acc1, acc2, acc3);
    spmm_step4<false>(x, vals, row, col, e + 4, e_end, row_base, kb, fn, acc0, acc1, acc2, acc3);
  }
  // Tail: at most 2 guarded steps.
  for (; e < e_end; e += 4) {
    spmm_step4<true>(x, vals, row, col, e, e_end, row_base, kb, fn, acc0, acc1, acc2, acc3);
  }

  // ---- normalize by degree and store: per v the wave writes two contiguous
  //      64B runs (rows v and v+8), coalesced; 32-bit offsets again. ----
  #pragma unroll
  for (int v = 0; v < 8; ++v) {
    const int m = v + (half << 3);
    const float rd = __shfl(rdeg, m, 32);   // per-lane src lane -> ds_bpermute
    const unsigned o = (unsigned)(row_base + m) * 64u + (unsigned)fn;
    out[o]       = acc0[v] * rd;
    out[o + 16u] = acc1[v] * rd;
    out[o + 32u] = acc2[v] * rd;
    out[o + 48u] = acc3[v] * rd;
  }
}

extern "C" void kernel_launch(void* const* d_in, const int* in_sizes, int n_in,
                              void* d_out, int out_size, void* d_ws, size_t ws_size,
                              hipStream_t stream) {
  const float* x    = (const float*)d_in[0];   // [N, 64]
  const float* vals = (const float*)d_in[1];   // [E]
  const int*   row  = (const int*)d_in[2];     // [E] sorted
  const int*   col  = (const int*)d_in[3];     // [E]
  float* out = (float*)d_out;                  // [N, 64]

  const int n_edges = in_sizes[1];
  const int n_nodes = in_sizes[0] / 64;

  const int groups = (n_nodes + 15) / 16;                     // one wave per group
  const int blocks = (groups + WAVES_PER_BLOCK - 1) / WAVES_PER_BLOCK;

  spmm_mean_wmma_f32<<<blocks, BLOCK_THREADS, 0, stream>>>(
      x, vals, row, col, out, n_nodes, n_edges);
}